// GIB_Layer_9500467658969
// MI455X (gfx1250) — compile-verified
//
#include <hip/hip_runtime.h>

typedef __attribute__((ext_vector_type(2))) float v2f;
typedef __attribute__((ext_vector_type(8))) float v8f;

#define KSUP  32          // support points per query
#define OBS   16          // observers
#define EPSF  1e-8f
#define LOG2E 1.4426950408889634f
#define BIGX  1e30f       // folded reach-mask: x2' = BIGX when out of reach
#define WAVES_PER_BLOCK 4
#define QT    16          // queries per wave tile (matches WMMA M)
#define PITCH 34          // LDS row pitch in floats

static __device__ __forceinline__ float swap16(float v) {
    return __shfl_xor(v, 16, 32);   // exchange lane <-> lane^16
}

__global__ __launch_bounds__(128) void gib_layer_kernel(
    const float* __restrict__ points,
    const float* __restrict__ q_coords,
    const int*   __restrict__ support_idxs,
    const float* __restrict__ cy_radius,
    const float* __restrict__ disk_radius,
    const float* __restrict__ disk_width,
    const float* __restrict__ cone_radius,
    const float* __restrict__ cone_inc,
    const float* __restrict__ ellip_radii,
    const float* __restrict__ lambdas,
    float* __restrict__ out,
    int M)
{
    __shared__ float qtile[WAVES_PER_BLOCK][QT * PITCH];

    const int lane = threadIdx.x & 31;
    const int wave = threadIdx.x >> 5;
    const int tile = blockIdx.x * WAVES_PER_BLOCK + wave;
    const int m0 = tile * QT;
    if (m0 >= M) return;            // wave-uniform: EXEC stays all-1s below

    const bool hi  = (lane & 16) != 0;
    const int  nlo = lane & 15;

    // ---- per-lane (lane = GIB g) coefficients: arg = cx2*x2' + cy2*y2 + cz2*z2 + crz*(rxy*z) + cc
    // (log2e and the -1/2 folded in; cx2 < 0 for every family -> BIGX masking is safe)
    const int fam = lane >> 3;      // 0=cylinder 1=cone 2=disk 3=ellipsoid
    const int gi  = lane & 7;
    float cx2, cy2, cz2, crz, cc;
    if (fam == 0) {
        float r = cy_radius[gi];
        float a = -0.5f * LOG2E / (r * r + EPSF);
        cx2 = a; cy2 = a; cz2 = 0.f; crz = 0.f; cc = 0.f;
    } else if (fam == 1) {
        // (r_xy - z*inc)^2 = (xy2 + EPS) - 2*inc*(r_xy*z) + inc^2*z^2
        float r   = cone_radius[gi];
        float inc = cone_inc[gi];
        float d   = 0.5f * LOG2E / (r * r + EPSF);
        cx2 = -d; cy2 = -d; cz2 = -d * inc * inc; crz = 2.f * d * inc; cc = -d * EPSF;
    } else if (fam == 2) {
        float r = disk_radius[gi], w = disk_width[gi];
        float a = -0.5f * LOG2E / (r * r + EPSF);
        cx2 = a; cy2 = a; cz2 = -0.5f * LOG2E / (w * w + EPSF); crz = 0.f; cc = 0.f;
    } else {
        float rx = ellip_radii[gi * 3 + 0];
        float ry = ellip_radii[gi * 3 + 1];
        float rz = ellip_radii[gi * 3 + 2];
        cx2 = -0.5f * LOG2E / (rx * rx + EPSF);
        cy2 = -0.5f * LOG2E / (ry * ry + EPSF);
        cz2 = -0.5f * LOG2E / (rz * rz + EPSF);
        crz = 0.f; cc = 0.f;
    }

    // ---- build WMMA B operands (coef matrix, 4 feats x 32 g) + C init (cc), once ----
    // B 4x16 layout: lanes0-15 hold K=0,1 in vgpr0,1; lanes16-31 hold K=2,3.
    // features: K0=x2 coef, K1=y2 coef, K2=z2 coef, K3=rxyz coef; col N -> g = N + 16*h
    float cx2s = swap16(cx2), cy2s = swap16(cy2);
    float cz2s = swap16(cz2), crzs = swap16(crz), ccs = swap16(cc);
    v2f b_h0, b_h1;
    b_h0.x = hi ? cz2s : cx2;   b_h0.y = hi ? crzs : cy2;    // g-half 0 (g = nlo)
    b_h1.x = hi ? cz2  : cx2s;  b_h1.y = hi ? crz  : cy2s;   // g-half 1 (g = nlo+16)
    float cc_h0 = hi ? ccs : cc;
    float cc_h1 = hi ? cc  : ccs;
    v8f c0, c1;
    #pragma unroll
    for (int i = 0; i < 8; ++i) { c0[i] = cc_h0; c1[i] = cc_h1; }

    // ---- per-query: features (lane = neighbor k), ARG via WMMA, exp, row-sum ----
    for (int j = 0; j < QT; ++j) {
        int m = m0 + j;
        if (m >= M) m = M - 1;      // uniform clamp; extra rows never stored

        int   idx = support_idxs[m * KSUP + lane];
        float qx = q_coords[m * 3 + 0];
        float qy = q_coords[m * 3 + 1];
        float qz = q_coords[m * 3 + 2];
        float rx = points[idx * 3 + 0] - qx;
        float ry = points[idx * 3 + 1] - qy;
        float rz = points[idx * 3 + 2] - qz;
        float x2 = rx * rx, y2 = ry * ry, z2 = rz * rz;
        float xy2 = x2 + y2;
        float d2  = xy2 + z2;
        // raw v_sqrt_f32: arg >= 1e-8, never denormal/negative -> skip IEEE fixup
        float rxyz = __builtin_amdgcn_sqrtf(xy2 + EPSF) * rz;   // r_xy * z
        float x2m  = (d2 <= 1.0f) ? x2 : BIGX;                  // reach mask folded in

        // A tiles (16 rows x 4 feats): lane&15 = row; lanes16-31 hold feats 2,3
        float xs = swap16(x2m), ys = swap16(y2), zs = swap16(z2), rs = swap16(rxyz);
        v2f a0, a1;                                      // k-tiles 0 (k=0..15), 1 (k=16..31)
        a0.x = hi ? zs : x2m;   a0.y = hi ? rs   : y2;
        a1.x = hi ? z2 : xs;    a1.y = hi ? rxyz : ys;

        // ARG tiles: (k-tile) x (g-half), D = A*Coef + cc
        v8f d00 = __builtin_amdgcn_wmma_f32_16x16x4_f32(false, a0, false, b_h0, (short)0, c0, false, false);
        v8f d01 = __builtin_amdgcn_wmma_f32_16x16x4_f32(false, a0, false, b_h1, (short)0, c1, false, false);
        v8f d10 = __builtin_amdgcn_wmma_f32_16x16x4_f32(false, a1, false, b_h0, (short)0, c0, false, false);
        v8f d11 = __builtin_amdgcn_wmma_f32_16x16x4_f32(false, a1, false, b_h1, (short)0, c1, false, false);

        // w = exp2(arg); masked rows already -> 0. Sum over all 32 k (rows of D tiles).
        float s0 = 0.f, s1 = 0.f;
        #pragma unroll
        for (int v = 0; v < 8; ++v) {
            s0 += __builtin_amdgcn_exp2f(d00[v]);
            s0 += __builtin_amdgcn_exp2f(d10[v]);
            s1 += __builtin_amdgcn_exp2f(d01[v]);
            s1 += __builtin_amdgcn_exp2f(d11[v]);
        }
        // lanes<16 hold rows 0-7 partial, lanes>=16 rows 8-15 partial -> combine halves
        s0 += swap16(s0);
        s1 += swap16(s1);

        // Q[j][g] (mean over K): col g = nlo (+16); both lane halves write identical values
        qtile[wave][j * PITCH + nlo]      = s0 * (1.0f / KSUP);
        qtile[wave][j * PITCH + nlo + 16] = s1 * (1.0f / KSUP);
    }

    // ---- final WMMA chain: out_tile(16x16) = Q_tile(16x32) @ lambdas(32x16) ----
    v8f c = {};
    #pragma unroll
    for (int ch = 0; ch < 8; ++ch) {
        int g0 = ch * 4 + 2 * (hi ? 1 : 0);
        v2f a, b;
        a.x = qtile[wave][nlo * PITCH + g0 + 0];
        a.y = qtile[wave][nlo * PITCH + g0 + 1];
        b.x = lambdas[(g0 + 0) * OBS + nlo];
        b.y = lambdas[(g0 + 1) * OBS + nlo];
        c = __builtin_amdgcn_wmma_f32_16x16x4_f32(false, a, false, b, (short)0, c,
                                                  false, false);
    }

    // D layout: lane<16 -> row = v, col = lane; lane>=16 -> row = v+8, col = lane-16
    const int rbase = m0 + (hi ? 8 : 0);
    if (m0 + QT <= M) {
        // full tile (always the case when M % 16 == 0): straight-line stores
        #pragma unroll
        for (int v = 0; v < 8; ++v) {
            out[(rbase + v) * OBS + nlo] = c[v];
        }
    } else {
        #pragma unroll
        for (int v = 0; v < 8; ++v) {
            int row = rbase + v;
            if (row < M) out[row * OBS + nlo] = c[v];
        }
    }
}

extern "C" void kernel_launch(void* const* d_in, const int* in_sizes, int n_in,
                              void* d_out, int out_size, void* d_ws, size_t ws_size,
                              hipStream_t stream) {
    (void)n_in; (void)out_size; (void)d_ws; (void)ws_size;
    const float* points       = (const float*)d_in[0];
    const float* q_coords     = (const float*)d_in[1];
    const int*   support_idxs = (const int*)  d_in[2];
    const float* cy_radius    = (const float*)d_in[3];
    const float* disk_radius  = (const float*)d_in[4];
    const float* disk_width   = (const float*)d_in[5];
    const float* cone_radius  = (const float*)d_in[6];
    const float* cone_inc     = (const float*)d_in[7];
    const float* ellip_radii  = (const float*)d_in[8];
    const float* lambdas      = (const float*)d_in[9];
    float* out = (float*)d_out;

    const int M = in_sizes[1] / 3;                   // q_coords is (M,3)
    const int tiles  = (M + QT - 1) / QT;            // 16-query wave tiles
    const int blocks = (tiles + WAVES_PER_BLOCK - 1) / WAVES_PER_BLOCK;

    gib_layer_kernel<<<blocks, 32 * WAVES_PER_BLOCK, 0, stream>>>(
        points, q_coords, support_idxs, cy_radius, disk_radius, disk_width,
        cone_radius, cone_inc, ellip_radii, lambdas, out, M);
}